// AffinityLoss_74723841015958
// MI455X (gfx1250) — compile-verified
//
#include <hip/hip_runtime.h>

#define IMG_H 512
#define IMG_W 512
#define NB    8
#define HW    (IMG_H * IMG_W)          // 262144
#define NTOT  (NB * HW)                // 2097152
#define TILES_PER_IMG 1024             // (512/16)*(512/16)
#define NBLK  (NB * TILES_PER_IMG)     // 8192

typedef float v2f __attribute__((ext_vector_type(2)));
typedef float v8f __attribute__((ext_vector_type(8)));

// ---------------------------------------------------------------------------
// Wave32 sum via V_WMMA_F32_16X16X4_F32:
//   A[m][k]: vgpr0 -> K=0 (lanes 0-15) / K=2 (lanes 16-31); vgpr1 -> K=1/K=3.
//   Put x in vgpr0, 0 in vgpr1, B = all ones => D[m][n] = x[m] + x[m+16].
//   D vgpr r: lanes 0-15 hold row r, lanes 16-31 hold row r+8. Summing the 8
//   D vgprs per-lane gives lanes0-15 = sum(rows 0..7), lanes16-31 = sum(rows
//   8..15); one cross-half shuffle finishes the 32-lane sum (valid lanes 0-15).
// ---------------------------------------------------------------------------
__device__ __forceinline__ float wave_sum32(float x) {
#if __has_builtin(__builtin_amdgcn_wmma_f32_16x16x4_f32)
    v2f a; a[0] = x;    a[1] = 0.0f;
    v2f b; b[0] = 1.0f; b[1] = 1.0f;   // all-ones B: layout-independent
    v8f c = {};
    v8f d = __builtin_amdgcn_wmma_f32_16x16x4_f32(
        /*neg_a=*/false, a, /*neg_b=*/false, b,
        /*c_mod=*/(short)0, c, /*reuse_a=*/false, /*reuse_b=*/false);
    float s = d[0] + d[1] + d[2] + d[3] + d[4] + d[5] + d[6] + d[7];
    s += __shfl(s, 16, 32);            // lanes 0-15 now hold full wave sum
    return s;
#else
    for (int off = 16; off > 0; off >>= 1) x += __shfl_down(x, off, 32);
    return __shfl(x, 0, 32);
#endif
}

// Deterministic 256-thread block sum (8 waves). All threads return the sum.
__device__ __forceinline__ float block_sum256(float x, float* sbuf8) {
    float w = wave_sum32(x);
    unsigned tid = threadIdx.x;
    if ((tid & 31u) == 0u) sbuf8[tid >> 5] = w;   // lane 0 of each wave
    __syncthreads();
    float r = ((sbuf8[0] + sbuf8[1]) + (sbuf8[2] + sbuf8[3]))
            + ((sbuf8[4] + sbuf8[5]) + (sbuf8[6] + sbuf8[7]));
    __syncthreads();                               // allow sbuf reuse
    return r;
}

// ---------------------------------------------------------------------------
// 2 Euler steps (dx = sqrt(2)/5) then 10 steps with f/5; nearest sampling
// with round-half-to-even (rintf == jnp.round) and clamp (== jnp.clip).
// Returns endpoint after 2 steps (fp) and after all 12 (q).
// ---------------------------------------------------------------------------
__device__ __forceinline__ void advect12(const float* __restrict__ f,
                                         float py, float px,
                                         float& fpy, float& fpx,
                                         float& qy,  float& qx) {
    const float DX = 0.28284271247461906f;  // f32(sqrt(2)/5), JAX weak-type promotion
#pragma unroll
    for (int i = 0; i < 2; ++i) {
        float ry = fminf(fmaxf(rintf(py), 0.0f), 511.0f);
        float rx = fminf(fmaxf(rintf(px), 0.0f), 511.0f);
        int o = (int)ry * IMG_W + (int)rx;
        float vy = f[o], vx = f[HW + o];
        py += DX * vy;
        px += DX * vx;
    }
    fpy = py; fpx = px;
#pragma unroll
    for (int i = 0; i < 10; ++i) {
        float ry = fminf(fmaxf(rintf(py), 0.0f), 511.0f);
        float rx = fminf(fmaxf(rintf(px), 0.0f), 511.0f);
        int o = (int)ry * IMG_W + (int)rx;
        py += f[o]      / 5.0f;         // fs = f/5 sampled: same as sample/5
        px += f[HW + o] / 5.0f;
    }
    qy = py; qx = px;
}

// ---------------------------------------------------------------------------
// Fused kernel: one 16x16 tile per block. Each block advects its 18x18 halo
// (q-points kept in LDS, never spilled to global), computes endpoint-MSE
// contributions for interior pixels, then the 8-neighbor affinity tests from
// LDS. Per-block partial sums (lossA count, lossE, lossB) go to workspace.
// ---------------------------------------------------------------------------
__global__ __launch_bounds__(256) void affinity_main(
    const float* __restrict__ flow_pred, const float* __restrict__ dist_pred,
    const float* __restrict__ flow_gt,   const float* __restrict__ dist_gt,
    float* __restrict__ partA, float* __restrict__ partE, float* __restrict__ partB)
{
    __shared__ float s_qpy[18][19];
    __shared__ float s_qpx[18][19];
    __shared__ float s_qgy[18][19];
    __shared__ float s_qgx[18][19];
    __shared__ float s_fg [18][19];
    __shared__ float sred[8];

    const unsigned tid  = threadIdx.x;
    const unsigned blk  = blockIdx.x;
    const unsigned b    = blk >> 10;            // 1024 tiles per image
    const unsigned tile = blk & 1023u;
    const int ty0 = (int)((tile >> 5) << 4);
    const int tx0 = (int)((tile & 31u) << 4);

    const float* fp = flow_pred + (size_t)b * (2 * HW);
    const float* fg = flow_gt   + (size_t)b * (2 * HW);
    const float* dp = dist_pred + (size_t)b * HW;
    const float* dg = dist_gt   + (size_t)b * HW;

    float eAcc = 0.0f;

    for (unsigned i = tid; i < 18u * 18u; i += 256u) {
        int ly = (int)(i / 18u), lx = (int)(i % 18u);
        int gy = ty0 - 1 + ly,  gx = tx0 - 1 + lx;
        float qpy = 0.f, qpx = 0.f, qgy = 0.f, qgx = 0.f, fgv = 0.f;
        if (gy >= 0 && gy < IMG_H && gx >= 0 && gx < IMG_W) {
            int o = gy * IMG_W + gx;
            // fg shared between pred/gt graphs; OOB halo keeps fgv=0 which
            // reproduces the reference's `valid` mask (conn needs fgn&&valid).
            fgv = (dp[o] >= 0.0f || dg[o] >= 0.0f) ? 1.0f : 0.0f;
            float fppy, fppx, fgpy, fgpx;
            advect12(fp, (float)gy, (float)gx, fppy, fppx, qpy, qpx);
            advect12(fg, (float)gy, (float)gx, fgpy, fgpx, qgy, qgx);
            if (ly >= 1 && ly <= 16 && lx >= 1 && lx <= 16) {   // interior once
                float dey = fppy - fgpy, dex = fppx - fgpx;
                eAcc += dey * dey + dex * dex;
            }
        }
        s_qpy[ly][lx] = qpy; s_qpx[ly][lx] = qpx;
        s_qgy[ly][lx] = qgy; s_qgx[ly][lx] = qgx;
        s_fg [ly][lx] = fgv;
    }
    __syncthreads();

    const int ty = (int)(tid >> 4) + 1;
    const int tx = (int)(tid & 15u) + 1;
    const float fgc = s_fg[ty][tx];
    const float cpy = s_qpy[ty][tx], cpx = s_qpx[ty][tx];
    const float cgy = s_qgy[ty][tx], cgx = s_qgx[ty][tx];

    // 8 non-center offsets; center has thr=0 => conn always false, contributes 0.
    const int SY[8] = {-1,-1,-1, 0, 0, 1, 1, 1};
    const int SX[8] = {-1, 0, 1,-1, 1,-1, 0, 1};

    int mism = 0, csp = 0, csg = 0;
#pragma unroll
    for (int k = 0; k < 8; ++k) {
        int sy = SY[k], sx = SX[k];
        float thr = (float)(sy * sy + sx * sx);
        int ny = ty + sy, nx = tx + sx;
        float fgn = s_fg[ny][nx];
        float d1 = cpy - s_qpy[ny][nx];
        float d2 = cpx - s_qpx[ny][nx];
        float e1 = cgy - s_qgy[ny][nx];
        float e2 = cgx - s_qgx[ny][nx];
        bool common = (fgc > 0.5f) && (fgn > 0.5f);
        int connp = ((d1 * d1 + d2 * d2) < thr) && common;
        int conng = ((e1 * e1 + e2 * e2) < thr) && common;
        mism += (connp != conng);   // (ag_p - ag_g)^2 on {0,1} == mismatch
        csp  += connp;
        csg  += conng;
    }
    float la = (float)mism;
    bool bdp = (csp >= 2) && (csp < 8);   // csum >= DIM && csum < 3^DIM - 1
    bool bdg = (csg >= 2) && (csg < 8);
    // BCE with p,t in {0,1} and log clamp at -100  =>  100 per mismatch
    float lb = (bdp != bdg) ? 100.0f : 0.0f;

    float rE = block_sum256(eAcc, sred);
    float rA = block_sum256(la,   sred);
    float rB = block_sum256(lb,   sred);
    if (tid == 0) { partA[blk] = rA; partE[blk] = rE; partB[blk] = rB; }
}

// Deterministic final reduction of the 8192 per-block partials.
__global__ __launch_bounds__(256) void finalize_k(
    const float* __restrict__ pA, const float* __restrict__ pE,
    const float* __restrict__ pB, float* __restrict__ out)
{
    __shared__ float sA[256], sE[256], sB[256];
    unsigned t = threadIdx.x;
    float a = 0.f, e = 0.f, bb = 0.f;
    for (unsigned i = t; i < NBLK; i += 256u) { a += pA[i]; e += pE[i]; bb += pB[i]; }
    sA[t] = a; sE[t] = e; sB[t] = bb;
    __syncthreads();
    for (unsigned st = 128; st > 0; st >>= 1) {
        if (t < st) { sA[t] += sA[t + st]; sE[t] += sE[t + st]; sB[t] += sB[t + st]; }
        __syncthreads();
    }
    if (t == 0) {
        out[0] = sA[0] / (9.0f * (float)NTOT);   // lossA: mean over [B,9,H,W]
        out[1] = sE[0] / (2.0f * (float)NTOT);   // lossE: mean over [B,2,H,W]
        out[2] = sB[0] / (float)NTOT;            // lossB: mean over [B,H,W]
    }
}

extern "C" void kernel_launch(void* const* d_in, const int* in_sizes, int n_in,
                              void* d_out, int out_size, void* d_ws, size_t ws_size,
                              hipStream_t stream) {
    const float* flow_pred = (const float*)d_in[0];   // [8,2,512,512]
    const float* dist_pred = (const float*)d_in[1];   // [8,512,512]
    const float* flow_gt   = (const float*)d_in[2];   // [8,2,512,512]
    const float* dist_gt   = (const float*)d_in[3];   // [8,512,512]

    float* ws    = (float*)d_ws;                      // 3*8192 floats = 96 KB
    float* partA = ws;
    float* partE = ws + NBLK;
    float* partB = ws + 2 * NBLK;

    affinity_main<<<NBLK, 256, 0, stream>>>(flow_pred, dist_pred, flow_gt, dist_gt,
                                            partA, partE, partB);
    finalize_k<<<1, 256, 0, stream>>>(partA, partE, partB, (float*)d_out);
}